// MotionLayer_47485158425272
// MI455X (gfx1250) — compile-verified
//
#include <hip/hip_runtime.h>
#include <math.h>

typedef __attribute__((ext_vector_type(2))) float v2f;
typedef __attribute__((ext_vector_type(8))) float v8f;

#define B_ 16
#define C_ 3
#define T_ 16
#define Td_ 15
#define H_ 224
#define W_ 224
#define HW_ 50176
#define NFRM 240   // B_*Td_
#define NPART 1024

struct MParams {
  float s, bias;
  int w_klo, w_khi; float w_wl, w_wh;   // width window  (smooth ratio_h along W)
  int h_klo, h_khi; float h_wl, h_wh;   // height window (smooth ratio_w along H)
  float Wt[Td_ * 16];                   // temporal weights, K padded to 16 (col 15 = 0)
};

__device__ __forceinline__ float recip_auto(float p, float bound) {
  float slope = (bound <= 32.f) ? 10.f : 100.f;
  return (1.f - bound) / (1.f + expf(-slope * p)) + bound;
}

__device__ __forceinline__ void closest_odd(float num, int& klo, int& khi, float& wl, float& wh) {
  num = rintf(num * 1000.f) / 1000.f;     // rintf = round-half-even, matches jnp.round
  float base = floorf(num);
  float lower = (fmodf(base, 2.f) != 0.f) ? base : base - 1.f;
  klo = (int)lower;
  khi = (int)(lower + 2.f);
  float hw = (num - lower) * 0.5f;
  wl = 1.f - hw; wh = hw;
}

__global__ void setup_kernel(const float* h_in, const float* w_in, const float* t_in,
                             const float* m_in, const float* nn_in,
                             MParams* P, int* nz) {
  int tid = threadIdx.x;
  if (tid < NFRM) nz[tid] = 0;
  if (tid == 0) {
    P->s = 5.f / (0.45f * fabsf(tanhf(m_in[0])) + 0.1f);
    P->bias = 0.6f * tanhf(nn_in[0]);
    float ww = recip_auto(w_in[0], (float)W_);
    closest_odd(ww, P->w_klo, P->w_khi, P->w_wl, P->w_wh);
    float hh = recip_auto(h_in[0], (float)H_);
    closest_odd(hh, P->h_klo, P->h_khi, P->h_wl, P->h_wh);
    float tw = recip_auto(t_in[0], (float)Td_);
    int klo, khi; float wl, wh;
    closest_odd(tw, klo, khi, wl, wh);
    int plo = (klo - 1) >> 1, phi = (khi - 1) >> 1;
    for (int t = 0; t < Td_; ++t) {
      for (int k = 0; k < 16; ++k) {
        float wgt = 0.f;
        if (k < Td_) {
          int lo = t - plo; if (lo < 0) lo = 0;
          int hi = t + plo; if (hi > Td_ - 1) hi = Td_ - 1;
          if (k >= lo && k <= hi) wgt += wl / (float)klo;
          lo = t - phi; if (lo < 0) lo = 0;
          hi = t + phi; if (hi > Td_ - 1) hi = Td_ - 1;
          if (k >= lo && k <= hi) wgt += wh / (float)khi;
        }
        P->Wt[t * 16 + k] = wgt;
      }
    }
  }
}

// fd = gray[t+1]-gray[t]; also count nonzeros per frame (for static-frame fix)
__global__ void fd_kernel(const float* __restrict__ vid, float* __restrict__ fd,
                          int* __restrict__ nz) {
  int bid = blockIdx.x;
  int frame = bid / 196;
  int pos = (bid % 196) * 256 + threadIdx.x;   // HW_ = 196*256 exactly
  int b = frame / Td_, t = frame % Td_;
  const float* vb = vid + (size_t)b * C_ * T_ * HW_;
  const float GR[3] = {0.299f, 0.587f, 0.114f};
  const float SD[3] = {0.229f, 0.224f, 0.225f};
  const float MN[3] = {0.485f, 0.456f, 0.406f};
  float g0 = 0.f, g1 = 0.f;
#pragma unroll
  for (int c = 0; c < 3; ++c) {
    float x0 = vb[((size_t)c * T_ + t) * HW_ + pos];
    float x1 = vb[((size_t)c * T_ + t + 1) * HW_ + pos];
    g0 += GR[c] * (x0 * SD[c] + MN[c]);
    g1 += GR[c] * (x1 * SD[c] + MN[c]);
  }
  float d = g1 - g0;
  fd[(size_t)frame * HW_ + pos] = d;
  __shared__ int red[256];
  red[threadIdx.x] = (d != 0.f) ? 1 : 0;
  __syncthreads();
  for (int s = 128; s > 0; s >>= 1) {
    if (threadIdx.x < s) red[threadIdx.x] += red[threadIdx.x + s];
    __syncthreads();
  }
  if (threadIdx.x == 0 && red[0] != 0) atomicAdd(&nz[frame], red[0]);
}

__global__ void src_kernel(const int* __restrict__ nz, int* __restrict__ src) {
  int b = threadIdx.x;
  if (b < B_) {
    int cur = 0;
    src[b * Td_] = 0;
    for (int j = 1; j < Td_; ++j) {
      if (nz[b * Td_ + j] != 0) cur = j;
      src[b * Td_ + j] = cur;
    }
  }
}

// ratio_h: sum/count over H per (frame, w) — coalesced column walks
__global__ void colsum_kernel(const float* __restrict__ fd, const int* __restrict__ src,
                              float* __restrict__ ratio_h) {
  int idx = blockIdx.x * 256 + threadIdx.x;
  if (idx >= NFRM * W_) return;
  int frame = idx / W_, w = idx % W_;
  int b = frame / Td_;
  int sf = b * Td_ + src[frame];
  const float* f = fd + (size_t)sf * HW_ + w;
  float sum = 0.f; int cnt = 0;
  for (int h = 0; h < H_; ++h) {
    float v = f[(size_t)h * W_];
    sum += v; cnt += (v != 0.f);
  }
  ratio_h[(size_t)frame * W_ + w] = sum / ((float)cnt + 1e-6f);
}

// ratio_w: one wave per (frame, h) row, shuffle reduction
__global__ void rowsum_kernel(const float* __restrict__ fd, const int* __restrict__ src,
                              float* __restrict__ ratio_w) {
  int gw = (blockIdx.x * 256 + threadIdx.x) >> 5;
  int lane = threadIdx.x & 31;
  if (gw >= NFRM * H_) return;
  int frame = gw / H_, h = gw % H_;
  int b = frame / Td_;
  int sf = b * Td_ + src[frame];
  const float* row = fd + (size_t)sf * HW_ + (size_t)h * W_;
  float sum = 0.f; int cnt = 0;
  for (int w = lane; w < W_; w += 32) {
    float v = row[w];
    sum += v; cnt += (v != 0.f);
  }
#pragma unroll
  for (int off = 16; off > 0; off >>= 1) {
    sum += __shfl_xor(sum, off, 32);
    cnt += __shfl_xor(cnt, off, 32);
  }
  if (lane == 0) ratio_w[(size_t)frame * H_ + h] = sum / ((float)cnt + 1e-6f);
}

// dual-odd-window box smoothing with replicate padding, via LDS prefix scan.
// Output is padded to 16 t-rows per b (row 15 zeroed for the WMMA K-pad).
__global__ void smooth_kernel(const float* __restrict__ in, float* __restrict__ out,
                              const MParams* __restrict__ P, int useWidth) {
  int bt = blockIdx.x;              // b*16 + t
  int t = bt & 15;
  int b = bt >> 4;
  int tid = threadIdx.x;
  float* orow = out + (size_t)bt * W_;       // H_ == W_ == 224
  if (t == Td_) {
    if (tid < W_) orow[tid] = 0.f;
    return;
  }
  const float* irow = in + ((size_t)b * Td_ + t) * W_;
  __shared__ float x[W_];
  __shared__ float sc[256];
  float xi = (tid < W_) ? irow[tid] : 0.f;
  if (tid < W_) x[tid] = xi;
  sc[tid] = (tid < W_) ? xi : 0.f;
  __syncthreads();
  for (int off = 1; off < 256; off <<= 1) {   // Hillis-Steele inclusive scan
    float tv = (tid >= off) ? sc[tid - off] : 0.f;
    __syncthreads();
    sc[tid] += tv;
    __syncthreads();
  }
  if (tid < W_) {
    int klo = useWidth ? P->w_klo : P->h_klo;
    int khi = useWidth ? P->w_khi : P->h_khi;
    float wl = useWidth ? P->w_wl : P->h_wl;
    float wh = useWidth ? P->w_wh : P->h_wh;
    float x0 = x[0], xL = x[W_ - 1];
    int ks[2] = {klo, khi};
    float wq[2] = {wl, wh};
    float r = 0.f;
#pragma unroll
    for (int q = 0; q < 2; ++q) {
      int k = ks[q];
      int p = (k - 1) >> 1;
      int hi = tid + p; if (hi > W_ - 1) hi = W_ - 1;
      int lo = tid - p; if (lo < 0) lo = 0;
      float s = sc[hi] - (lo > 0 ? sc[lo - 1] : 0.f);
      int e0 = p - tid;            if (e0 < 0) e0 = 0;
      int e1 = tid + p - (W_ - 1); if (e1 < 0) e1 = 0;
      s += (float)e0 * x0 + (float)e1 * xL;    // replicate padding
      r += wq[q] * s / (float)k;
    }
    orow[tid] = r;
  }
}

// One wave per 16x16 tile: G = Wt-weighted sum of outer products via 4x WMMA f32 16x16x4,
// then fuse sigmoid gate, full_att store, and 3-channel output.
__global__ void grid_att_out_kernel(const float* __restrict__ vid,
                                    const float* __restrict__ fd,
                                    const float* __restrict__ sm_w,
                                    const float* __restrict__ sm_h,
                                    const int* __restrict__ src,
                                    const MParams* __restrict__ P,
                                    float* __restrict__ fa,
                                    float* __restrict__ out) {
  int bid = blockIdx.x;
  int tile = bid % 196;
  int bt = bid / 196;
  int t = bt % Td_;
  int b = bt / Td_;
  int th = tile / 14, tw = tile % 14;
  int h0 = th * 16, w0 = tw * 16;
  int lane = threadIdx.x;
  int m = lane & 15;
  int half = lane >> 4;
  const float* Wtrow = P->Wt + t * 16;
  const float* smw = sm_w + (size_t)b * 16 * H_;  // [k][h], k padded to 16
  const float* smh = sm_h + (size_t)b * 16 * W_;  // [k][w]
  v8f c = {0.f, 0.f, 0.f, 0.f, 0.f, 0.f, 0.f, 0.f};
#pragma unroll
  for (int kk = 0; kk < 4; ++kk) {
    int kb = kk * 4 + half * 2;                   // lanes 0-15: K=0,1; lanes 16-31: K=2,3
    v2f a, bb;
    a.x = Wtrow[kb]     * smw[(size_t)kb * H_ + h0 + m];
    a.y = Wtrow[kb + 1] * smw[(size_t)(kb + 1) * H_ + h0 + m];
    bb.x = smh[(size_t)kb * W_ + w0 + m];
    bb.y = smh[(size_t)(kb + 1) * W_ + w0 + m];
    c = __builtin_amdgcn_wmma_f32_16x16x4_f32(false, a, false, bb, (short)0, c, false, false);
  }
  int sf = b * Td_ + src[bt];
  const float* fdf = fd + (size_t)sf * HW_;
  float* faf = fa + (size_t)bt * HW_;
  float s = P->s, bias = P->bias;
#pragma unroll
  for (int i = 0; i < 8; ++i) {
    int hh = h0 + i + half * 8;                   // C layout: M = i + 8*(lane>=16)
    int ww = w0 + m;
    size_t off = (size_t)hh * W_ + ww;
    float fdv = fdf[off];
    float att = 1.f / (1.f + expf(-s * (fdv - bias)));
    float v = att * c[i];
    faf[off] = v;
#pragma unroll
    for (int ch = 0; ch < 3; ++ch) {
      out[(((size_t)b * C_ + ch) * Td_ + t) * HW_ + off] =
          v * vid[(((size_t)b * C_ + ch) * T_ + (t + 1)) * HW_ + off];
    }
  }
}

__global__ void loss_partial_kernel(const float* __restrict__ fa, float* __restrict__ partials) {
  const long N = (long)B_ * (Td_ - 1) * HW_;
  long stride = (long)gridDim.x * blockDim.x;
  float acc = 0.f;
  for (long e = (long)blockIdx.x * blockDim.x + threadIdx.x; e < N; e += stride) {
    long bt14 = e / HW_;
    long pos = e % HW_;
    long b = bt14 / (Td_ - 1);
    long t = bt14 % (Td_ - 1);
    long i0 = (b * Td_ + t) * HW_ + pos;
    float d = fa[i0 + HW_] - fa[i0];
    acc += d * d;
  }
  __shared__ float red[256];
  red[threadIdx.x] = acc;
  __syncthreads();
  for (int s = 128; s > 0; s >>= 1) {
    if (threadIdx.x < s) red[threadIdx.x] += red[threadIdx.x + s];
    __syncthreads();
  }
  if (threadIdx.x == 0) partials[blockIdx.x] = red[0];
}

__global__ void loss_final_kernel(const float* __restrict__ partials, float* __restrict__ loss_out) {
  __shared__ float red[256];
  float a = partials[threadIdx.x] + partials[threadIdx.x + 256] +
            partials[threadIdx.x + 512] + partials[threadIdx.x + 768];
  red[threadIdx.x] = a;
  __syncthreads();
  for (int s = 128; s > 0; s >>= 1) {
    if (threadIdx.x < s) red[threadIdx.x] += red[threadIdx.x + s];
    __syncthreads();
  }
  if (threadIdx.x == 0) loss_out[0] = red[0] / (float)(T_ - 2);
}

extern "C" void kernel_launch(void* const* d_in, const int* in_sizes, int n_in,
                              void* d_out, int out_size, void* d_ws, size_t ws_size,
                              hipStream_t stream) {
  const float* vid = (const float*)d_in[0];
  const float* p_h = (const float*)d_in[1];
  const float* p_w = (const float*)d_in[2];
  const float* p_t = (const float*)d_in[3];
  const float* p_m = (const float*)d_in[4];
  const float* p_n = (const float*)d_in[5];
  float* out = (float*)d_out;
  (void)in_sizes; (void)n_in; (void)ws_size;

  char* p = (char*)d_ws;
  MParams* P = (MParams*)p;
  size_t ofs = 4096;
  float* fd       = (float*)(p + ofs); ofs += (size_t)NFRM * HW_ * 4;
  float* fa       = (float*)(p + ofs); ofs += (size_t)NFRM * HW_ * 4;
  float* ratio_h  = (float*)(p + ofs); ofs += (size_t)NFRM * W_ * 4;
  float* ratio_w  = (float*)(p + ofs); ofs += (size_t)NFRM * H_ * 4;
  float* sm_h     = (float*)(p + ofs); ofs += (size_t)B_ * 16 * W_ * 4;
  float* sm_w     = (float*)(p + ofs); ofs += (size_t)B_ * 16 * H_ * 4;
  float* partials = (float*)(p + ofs); ofs += (size_t)NPART * 4;
  int* nz         = (int*)(p + ofs);   ofs += (size_t)NFRM * 4;
  int* src        = (int*)(p + ofs);   ofs += (size_t)NFRM * 4;

  setup_kernel<<<1, 256, 0, stream>>>(p_h, p_w, p_t, p_m, p_n, P, nz);
  fd_kernel<<<NFRM * 196, 256, 0, stream>>>(vid, fd, nz);
  src_kernel<<<1, 32, 0, stream>>>(nz, src);
  colsum_kernel<<<(NFRM * W_) / 256, 256, 0, stream>>>(fd, src, ratio_h);
  rowsum_kernel<<<(NFRM * H_) / 8, 256, 0, stream>>>(fd, src, ratio_w);
  smooth_kernel<<<B_ * 16, 256, 0, stream>>>(ratio_h, sm_h, P, 1);
  smooth_kernel<<<B_ * 16, 256, 0, stream>>>(ratio_w, sm_w, P, 0);
  grid_att_out_kernel<<<NFRM * 196, 32, 0, stream>>>(vid, fd, sm_w, sm_h, src, P, fa, out);
  loss_partial_kernel<<<NPART, 256, 0, stream>>>(fa, partials);
  loss_final_kernel<<<1, 256, 0, stream>>>(partials, out + (size_t)out_size - 1);
}